// EncoderGPECls_21397527069063
// MI455X (gfx1250) — compile-verified
//
#include <hip/hip_runtime.h>
#include <cfloat>
#include <cmath>

typedef __attribute__((ext_vector_type(2))) float v2f;
typedef __attribute__((ext_vector_type(8))) float v8f;

#define BATCH 16
#define NPTS 2048
#define CINIT 64
#define KNN 32
#define SIGMA_ 0.3f
#define BN_EPS_ 1e-5f
#define STD_G 16   // partial-reduction groups for k_std

// -------------------------------------------------------------------------
// 1) init embed: h[b,n,c] = sum_d xyz[b,n,d] * W[c,d]  via V_WMMA_F32_16X16X4_F32
//    M = B*N = 32768 rows, Ncols = 64, K = 3 (padded to 4). One wave per tile.
//    Branch-free fragment build: load all 3 coords, select per wave-half.
// -------------------------------------------------------------------------
__global__ __launch_bounds__(256) void k_embed_wmma(const float* __restrict__ xyz,
                                                    const float* __restrict__ W,
                                                    float* __restrict__ h) {
  const int wid  = (blockIdx.x * 256 + (int)threadIdx.x) >> 5;  // 0..8191
  const int lane = threadIdx.x & 31;
  const int hf   = lane >> 4;       // wave half: K pair {0,1} vs {2,3}
  const int ln   = lane & 15;
  const int tileM = wid >> 2;       // 2048 tiles of 16 rows
  const int tileN = wid & 3;        // 4 tiles of 16 cols
  const int m  = tileM * 16 + ln;
  const int c  = tileN * 16 + ln;
  const float x0 = xyz[m * 3], x1 = xyz[m * 3 + 1], x2 = xyz[m * 3 + 2];
  const float w0 = W[c * 3],   w1 = W[c * 3 + 1],   w2 = W[c * 3 + 2];
  v2f a, b;
  a.x = hf ? x2 : x0;  a.y = hf ? 0.f : x1;     // A[m][K=2h], A[m][K=1+2h] (K=3 pad)
  b.x = hf ? w2 : w0;  b.y = hf ? 0.f : w1;     // B[k][n] = W[n][k]
  v8f d = {};
  d = __builtin_amdgcn_wmma_f32_16x16x4_f32(false, a, false, b, (short)0, d, false, false);
#pragma unroll
  for (int r = 0; r < 8; ++r) {
    int row = tileM * 16 + r + 8 * hf;          // C/D layout: VGPR r -> rows r, r+8
    h[row * CINIT + tileN * 16 + ln] = d[r];
  }
}

// -------------------------------------------------------------------------
// 2) BatchNorm batch-stats (biased var) + apply ReLU
// -------------------------------------------------------------------------
__global__ __launch_bounds__(256) void k_bn_stats(const float* __restrict__ h,
                                                  float* __restrict__ mu,
                                                  float* __restrict__ inv) {
  __shared__ float s1[256], s2[256];
  const int c = blockIdx.x;
  const int n = BATCH * NPTS;
  float a = 0.f, b = 0.f;
  for (int i = threadIdx.x; i < n; i += 256) {
    float v = h[i * CINIT + c];
    a += v; b += v * v;
  }
  s1[threadIdx.x] = a; s2[threadIdx.x] = b; __syncthreads();
  for (int o = 128; o > 0; o >>= 1) {
    if ((int)threadIdx.x < o) { s1[threadIdx.x] += s1[threadIdx.x + o]; s2[threadIdx.x] += s2[threadIdx.x + o]; }
    __syncthreads();
  }
  if (threadIdx.x == 0) {
    float m   = s1[0] / (float)n;
    float var = s2[0] / (float)n - m * m;
    mu[c]  = m;
    inv[c] = rsqrtf(var + BN_EPS_);
  }
}

__global__ void k_bn_apply(float* __restrict__ h, const float* __restrict__ mu,
                           const float* __restrict__ inv, const float* __restrict__ gamma,
                           const float* __restrict__ beta) {
  const int total = BATCH * NPTS * CINIT;
  for (int i = blockIdx.x * blockDim.x + threadIdx.x; i < total; i += gridDim.x * blockDim.x) {
    int c = i & (CINIT - 1);
    float v = (h[i] - mu[c]) * inv[c] * gamma[c] + beta[c];
    h[i] = fmaxf(v, 0.f);
  }
}

// -------------------------------------------------------------------------
// 3) Farthest point sampling: one 1024-thread block (32 wave32) per batch.
//    Coords staged in LDS once; running min-dists live in registers (<=2/thread);
//    per-iteration argmax via wave shuffles + one 32-entry LDS hop.
//    idx[0]=0, then argmax with first-max tie break (matches jnp.argmax).
// -------------------------------------------------------------------------
#define FPS_T 1024
__global__ __launch_bounds__(FPS_T) void k_fps(const float* __restrict__ xyz, int N, int S,
                                               int* __restrict__ fps_idx) {
  __shared__ float px[NPTS], py[NPTS], pz[NPTS];
  __shared__ float wv[32];
  __shared__ int   wi[32];
  __shared__ int   far_s;
  const int b = blockIdx.x;
  const float* p = xyz + (size_t)b * N * 3;
  const int tid = threadIdx.x;
  for (int i = tid; i < N; i += FPS_T) { px[i] = p[i*3]; py[i] = p[i*3+1]; pz[i] = p[i*3+2]; }
  float d0 = 1e10f, d1 = 1e10f;
  const int i0 = tid, i1 = tid + FPS_T;   // N <= 2*FPS_T
  if (tid == 0) far_s = 0;
  __syncthreads();
  for (int it = 0; it < S; ++it) {
    const int far = far_s;
    if (tid == 0) fps_idx[b * S + it] = far;
    const float fx = px[far], fy = py[far], fz = pz[far];
    float best = -1.f; int bi = 0x7fffffff;
    if (i0 < N) {
      float dx = px[i0]-fx, dy = py[i0]-fy, dz = pz[i0]-fz;
      d0 = fminf(d0, dx*dx + dy*dy + dz*dz);
      best = d0; bi = i0;
    }
    if (i1 < N) {
      float dx = px[i1]-fx, dy = py[i1]-fy, dz = pz[i1]-fz;
      d1 = fminf(d1, dx*dx + dy*dy + dz*dz);
      if (d1 > best) { best = d1; bi = i1; }   // i0 < i1: strict > keeps first max
    }
#pragma unroll
    for (int o = 16; o > 0; o >>= 1) {
      float ov = __shfl_down(best, o, 32);
      int   oi = __shfl_down(bi,   o, 32);
      if (ov > best || (ov == best && oi < bi)) { best = ov; bi = oi; }
    }
    if ((tid & 31) == 0) { wv[tid >> 5] = best; wi[tid >> 5] = bi; }
    __syncthreads();
    if (tid < 32) {
      best = wv[tid]; bi = wi[tid];
#pragma unroll
      for (int o = 16; o > 0; o >>= 1) {
        float ov = __shfl_down(best, o, 32);
        int   oi = __shfl_down(bi,   o, 32);
        if (ov > best || (ov == best && oi < bi)) { best = ov; bi = oi; }
      }
      if (tid == 0) far_s = bi;
    }
    __syncthreads();
  }
}

// -------------------------------------------------------------------------
// 4) gather sampled xyz / feat rows
// -------------------------------------------------------------------------
__global__ void k_gather(const float* __restrict__ xyz, const float* __restrict__ feat,
                         const int* __restrict__ fps_idx, int N, int S, int C,
                         float* __restrict__ xyz_s, float* __restrict__ feat_s) {
  const int b = blockIdx.x / S, s = blockIdx.x % S;
  const int idx = fps_idx[b * S + s];
  if (threadIdx.x < 3)
    xyz_s[(size_t)(b * S + s) * 3 + threadIdx.x] = xyz[(size_t)(b * N + idx) * 3 + threadIdx.x];
  for (int j = threadIdx.x; j < C; j += blockDim.x)
    feat_s[(size_t)(b * S + s) * C + j] = feat[(size_t)(b * N + idx) * C + j];
}

// -------------------------------------------------------------------------
// 5) kNN distance strip for one batch via WMMA f32 16x16x4:
//    dist[s,n] = |xs[s]|^2 + |x[n]|^2 - 2 * (xs[s] . x[n])
//    Block covers 16 rows x 128 cols (8 waves, one 16x16 WMMA tile each).
// -------------------------------------------------------------------------
__global__ __launch_bounds__(256) void k_knn_dist_wmma(const float* __restrict__ xs,
                                                       const float* __restrict__ x,
                                                       int S, int N, float* __restrict__ dist) {
  __shared__ float sN[16];
  const int mbase = blockIdx.y * 16;
  const int w    = threadIdx.x >> 5;
  const int lane = threadIdx.x & 31;
  const int hf   = lane >> 4;
  const int ln   = lane & 15;
  if (threadIdx.x < 16) {
    float a0 = xs[(mbase + threadIdx.x) * 3];
    float a1 = xs[(mbase + threadIdx.x) * 3 + 1];
    float a2 = xs[(mbase + threadIdx.x) * 3 + 2];
    sN[threadIdx.x] = a0 * a0 + a1 * a1 + a2 * a2;
  }
  __syncthreads();
  const int m = mbase + ln;
  const float s0 = xs[m * 3], s1 = xs[m * 3 + 1], s2 = xs[m * 3 + 2];
  const int n = blockIdx.x * 128 + w * 16 + ln;
  const float x0 = x[n * 3], x1 = x[n * 3 + 1], x2 = x[n * 3 + 2];
  const float nn = x0 * x0 + x1 * x1 + x2 * x2;
  v2f a, b;
  a.x = hf ? s2 : s0;  a.y = hf ? 0.f : s1;   // branchless K-pair select
  b.x = hf ? x2 : x0;  b.y = hf ? 0.f : x1;
  v8f d = {};
  d = __builtin_amdgcn_wmma_f32_16x16x4_f32(false, a, false, b, (short)0, d, false, false);
#pragma unroll
  for (int r = 0; r < 8; ++r) {
    int row = mbase + r + 8 * hf;
    dist[(size_t)row * N + n] = sN[r + 8 * hf] + nn - 2.f * d[r];
  }
}

// -------------------------------------------------------------------------
// 6) top-K (K=32) smallest-distance selection per row; first-index tie break
// -------------------------------------------------------------------------
__global__ __launch_bounds__(256) void k_knn_select(const float* __restrict__ dist, int N,
                                                    int* __restrict__ knn_idx) {
  __shared__ float sv[NPTS];
  __shared__ float rv[256];
  __shared__ int   ri[256];
  const int s = blockIdx.x;
  const float* row = dist + (size_t)s * N;
  for (int i = threadIdx.x; i < N; i += 256) sv[i] = row[i];
  __syncthreads();
  for (int k = 0; k < KNN; ++k) {
    float best = FLT_MAX; int bi = 0x7fffffff;
    for (int i = threadIdx.x; i < N; i += 256) {
      float v = sv[i];
      if (v < best) { best = v; bi = i; }
    }
    rv[threadIdx.x] = best; ri[threadIdx.x] = bi;
    __syncthreads();
    for (int o = 128; o > 0; o >>= 1) {
      if ((int)threadIdx.x < o) {
        float vo = rv[threadIdx.x + o];
        if (vo < rv[threadIdx.x] || (vo == rv[threadIdx.x] && ri[threadIdx.x + o] < ri[threadIdx.x])) {
          rv[threadIdx.x] = vo; ri[threadIdx.x] = ri[threadIdx.x + o];
        }
      }
      __syncthreads();
    }
    if (threadIdx.x == 0) { knn_idx[s * KNN + k] = ri[0]; sv[ri[0]] = FLT_MAX; }
    __syncthreads();
  }
}

// -------------------------------------------------------------------------
// 7a) per-k std partial sums over a slice of (b,s): xyz and feat in one pass
//     grid = (KNN, STD_G); writes [k][g][4] = {sum_x, sq_x, sum_f, sq_f}
// -------------------------------------------------------------------------
__global__ __launch_bounds__(256) void k_std_partial(const float* __restrict__ xyz,
                                                     const float* __restrict__ feat,
                                                     const float* __restrict__ xyz_s,
                                                     const float* __restrict__ feat_s,
                                                     const int* __restrict__ knn_idx,
                                                     int N, int S, int C,
                                                     float* __restrict__ part) {
  __shared__ float s1[256], s2[256], s3[256], s4[256];
  const int k = blockIdx.x, g = blockIdx.y;
  float ax = 0.f, qx = 0.f, af = 0.f, qf = 0.f;
  for (int p = g * 256 + threadIdx.x; p < BATCH * S; p += STD_G * 256) {
    int b = p / S, s = p % S;
    int idx = knn_idx[(size_t)(b * S + s) * KNN + k];
    for (int c = 0; c < 3; ++c) {
      float d = xyz[(size_t)(b * N + idx) * 3 + c] - xyz_s[(size_t)(b * S + s) * 3 + c];
      ax += d; qx += d * d;
    }
    const float* fr = feat   + (size_t)(b * N + idx) * C;
    const float* fc = feat_s + (size_t)(b * S + s) * C;
    for (int j = 0; j < C; ++j) { float d = fr[j] - fc[j]; af += d; qf += d * d; }
  }
  s1[threadIdx.x] = ax; s2[threadIdx.x] = qx; s3[threadIdx.x] = af; s4[threadIdx.x] = qf;
  __syncthreads();
  for (int o = 128; o > 0; o >>= 1) {
    if ((int)threadIdx.x < o) {
      s1[threadIdx.x] += s1[threadIdx.x + o]; s2[threadIdx.x] += s2[threadIdx.x + o];
      s3[threadIdx.x] += s3[threadIdx.x + o]; s4[threadIdx.x] += s4[threadIdx.x + o];
    }
    __syncthreads();
  }
  if (threadIdx.x == 0) {
    float* q = part + ((size_t)k * STD_G + g) * 4;
    q[0] = s1[0]; q[1] = s2[0]; q[2] = s3[0]; q[3] = s4[0];
  }
}

// 7b) finalize: std over n elems, ddof=1, clamp >= 1e-5
__global__ void k_std_final(const float* __restrict__ part, int S, int C,
                            float* __restrict__ std_xyz, float* __restrict__ std_feat) {
  const int k = threadIdx.x;
  if (k >= KNN) return;
  float ax = 0.f, qx = 0.f, af = 0.f, qf = 0.f;
  for (int g = 0; g < STD_G; ++g) {
    const float* q = part + ((size_t)k * STD_G + g) * 4;
    ax += q[0]; qx += q[1]; af += q[2]; qf += q[3];
  }
  {
    double n = (double)BATCH * S * 3;
    double var = ((double)qx - (double)ax * ax / n) / (n - 1.0);
    std_xyz[k] = fmaxf(sqrtf(fmaxf((float)var, 0.f)), 1e-5f);
  }
  {
    double n = (double)BATCH * S * C;
    double var = ((double)qf - (double)af * af / n) / (n - 1.0);
    std_feat[k] = fmaxf(sqrtf(fmaxf((float)var, 0.f)), 1e-5f);
  }
}

// -------------------------------------------------------------------------
// 8) fused stage body: normalize, concat, GPE, fw=(f+pe)*pe, mean+max over K, GELU
//    feat gathers go straight to global (feat tensor is 8 MB -> L2-resident);
//    row starts prefetched (global_prefetch_b8) to hide first-touch latency.
// -------------------------------------------------------------------------
__global__ __launch_bounds__(256) void k_stage(const float* __restrict__ xyz,
                                               const float* __restrict__ feat,
                                               const float* __restrict__ xyz_s,
                                               const float* __restrict__ feat_s,
                                               const int* __restrict__ knn_idx,
                                               const float* __restrict__ std_xyz,
                                               const float* __restrict__ std_feat,
                                               int N, int S, int C,
                                               float* __restrict__ out) {
  __shared__ int   kidx[KNN];
  __shared__ float sfe[KNN];
  __shared__ float xkn[KNN][3];
  __shared__ float fs[512];
  const int b = blockIdx.x / S, s = blockIdx.x % S;
  if (threadIdx.x < KNN) {
    kidx[threadIdx.x] = knn_idx[(size_t)(b * S + s) * KNN + threadIdx.x];
    sfe[threadIdx.x]  = std_feat[threadIdx.x];
  }
  __syncthreads();
  if (threadIdx.x < KNN)   // warm L2/L0 for the K gathered feature rows
    __builtin_prefetch(feat + (size_t)(b * N + kidx[threadIdx.x]) * C, 0, 3);
  if (threadIdx.x < KNN * 3) {
    int k = threadIdx.x / 3, c = threadIdx.x % 3;
    xkn[k][c] = (xyz[(size_t)(b * N + kidx[k]) * 3 + c] - xyz_s[(size_t)(b * S + s) * 3 + c]) / std_xyz[k];
  }
  for (int j = threadIdx.x; j < C; j += 256) fs[j] = feat_s[(size_t)(b * S + s) * C + j];
  __syncthreads();
  const int C2 = 2 * C;
  const int fd = (C2 + 2) / 3;           // ceil(C2/3)
  const int M3 = 3 * fd;
  const float inv_sig = 1.f / SIGMA_;
  for (int j = threadIdx.x; j < C2; j += 256) {
    int src = (int)(((long long)j * (M3 - 1)) / (C2 - 1));  // np.linspace + trunc
    int cc = src / fd, f = src % fd;
    float fv = -1.f + 2.f * (float)(f + 1) / (float)(fd + 1);
    float acc = 0.f, mx = -FLT_MAX;
    for (int k = 0; k < KNN; ++k) {
      float t  = (xkn[k][cc] - fv) * inv_sig;
      float pe = expf(-0.5f * t * t);
      float fk;
      if (j < C) fk = (feat[(size_t)(b * N + kidx[k]) * C + j] - fs[j]) / sfe[k];
      else       fk = fs[j - C];
      float fw = (fk + pe) * pe;
      acc += fw; mx = fmaxf(mx, fw);
    }
    float xx = acc * (1.f / KNN) + mx;
    out[(size_t)(b * S + s) * C2 + j] = 0.5f * xx * (1.f + erff(xx * 0.70710678118654752f));
  }
}

// -------------------------------------------------------------------------
// 9) stage readout: [max over S ; mean over S] -> d_out slice
// -------------------------------------------------------------------------
__global__ void k_pool(const float* __restrict__ feat, int S, int C2,
                       float* __restrict__ out, int off) {
  const int b = blockIdx.x;
  for (int j = threadIdx.x; j < C2; j += blockDim.x) {
    float mx = -FLT_MAX, sm = 0.f;
    for (int s = 0; s < S; ++s) {
      float v = feat[(size_t)(b * S + s) * C2 + j];
      mx = fmaxf(mx, v); sm += v;
    }
    out[b * 3840 + off + j]      = mx;
    out[b * 3840 + off + C2 + j] = sm / (float)S;
  }
}

// -------------------------------------------------------------------------
extern "C" void kernel_launch(void* const* d_in, const int* in_sizes, int n_in,
                              void* d_out, int out_size, void* d_ws, size_t ws_size,
                              hipStream_t stream) {
  const float* xyz_in = (const float*)d_in[0];   // [16,2048,3]
  const float* W      = (const float*)d_in[1];   // [64,3]
  const float* gamma  = (const float*)d_in[2];   // [64]
  const float* beta   = (const float*)d_in[3];   // [64]
  float* out = (float*)d_out;                    // [16,3840]
  float* ws  = (float*)d_ws;

  size_t o = 0;
  float* featA  = ws + o; o += (size_t)BATCH * NPTS * CINIT;   // 2M floats
  float* featB  = ws + o; o += (size_t)BATCH * NPTS * CINIT;   // 2M
  float* xyzA   = ws + o; o += (size_t)BATCH * NPTS * 3;
  float* xyzB   = ws + o; o += (size_t)BATCH * NPTS * 3;
  float* feat_s = ws + o; o += (size_t)BATCH * 1024 * CINIT;   // B*S*C = 1M max
  float* dist   = ws + o; o += (size_t)1024 * 2048;            // per-batch strip, reused
  float* bn_mu  = ws + o; o += 64;
  float* bn_inv = ws + o; o += 64;
  float* stdx   = ws + o; o += 32;
  float* stdf   = ws + o; o += 32;
  float* stdp   = ws + o; o += (size_t)KNN * STD_G * 4;        // std partials
  int* fps_i = (int*)(ws + o); o += (size_t)BATCH * 1024;
  int* knn_i = (int*)(ws + o); o += (size_t)BATCH * 1024 * KNN;

  // init embed + BN + ReLU
  k_embed_wmma<<<1024, 256, 0, stream>>>(xyz_in, W, featA);
  k_bn_stats<<<CINIT, 256, 0, stream>>>(featA, bn_mu, bn_inv);
  k_bn_apply<<<2048, 256, 0, stream>>>(featA, bn_mu, bn_inv, gamma, beta);

  const float* xyz_cur = xyz_in;
  float* feat_cur = featA; float* feat_nxt = featB;
  float* xs_buf = xyzA;    float* xs_other = xyzB;
  int N = NPTS, C = CINIT, off = 0;

  for (int st = 0; st < 4; ++st) {
    const int S = N / 2, C2 = 2 * C;
    k_fps<<<BATCH, FPS_T, 0, stream>>>(xyz_cur, N, S, fps_i);
    k_gather<<<BATCH * S, 256, 0, stream>>>(xyz_cur, feat_cur, fps_i, N, S, C, xs_buf, feat_s);
    for (int b = 0; b < BATCH; ++b) {
      dim3 g(N / 128, S / 16);
      k_knn_dist_wmma<<<g, 256, 0, stream>>>(xs_buf + (size_t)b * S * 3,
                                             xyz_cur + (size_t)b * N * 3, S, N, dist);
      k_knn_select<<<S, 256, 0, stream>>>(dist, N, knn_i + (size_t)b * S * KNN);
    }
    {
      dim3 g(KNN, STD_G);
      k_std_partial<<<g, 256, 0, stream>>>(xyz_cur, feat_cur, xs_buf, feat_s, knn_i, N, S, C, stdp);
      k_std_final<<<1, KNN, 0, stream>>>(stdp, S, C, stdx, stdf);
    }
    k_stage<<<BATCH * S, 256, 0, stream>>>(xyz_cur, feat_cur, xs_buf, feat_s, knn_i,
                                           stdx, stdf, N, S, C, feat_nxt);
    k_pool<<<BATCH, 256, 0, stream>>>(feat_nxt, S, C2, out, off);
    off += 2 * C2;
    // rotate buffers for next stage
    float* t = feat_cur; feat_cur = feat_nxt; feat_nxt = t;
    xyz_cur = xs_buf;
    float* u = xs_buf; xs_buf = xs_other; xs_other = u;
    N = S; C = C2;
  }
  (void)in_sizes; (void)n_in; (void)out_size; (void)ws_size;
}